// KGGraphConvolutionLayer_10943576670966
// MI455X (gfx1250) — compile-verified
//
#include <hip/hip_runtime.h>

#define N_ENT 100000
#define D     128
#define R_REL 8
#define E_EDGE 131072
#define B_HEAD 32768

typedef __attribute__((ext_vector_type(2))) float v2f;
typedef __attribute__((ext_vector_type(8))) float v8f;

// Guaranteed native non-returning fp32 global atomic add (no CAS loop).
__device__ inline void atomic_add_f32(float* addr, float val) {
    asm volatile("global_atomic_add_f32 %0, %1, off" :: "v"(addr), "v"(val) : "memory");
}

// ---------------------------------------------------------------------------
// Kernel 1: emb[n,o] = sum_i x[n,i] * W[o,i]   (x [N,128], W [128,128] row-major)
// One block = one 16-row M tile; 8 waves cover the 8 output 16-col tiles.
// fp32 WMMA 16x16x4, K-loop of 32 steps.
// ---------------------------------------------------------------------------
__global__ __launch_bounds__(256) void gemm_xWt(const float* __restrict__ x,
                                                const float* __restrict__ W,
                                                float* __restrict__ emb) {
    const int wave = threadIdx.x >> 5;        // 0..7 -> output col tile
    const int lane = threadIdx.x & 31;
    const int half = lane >> 4;               // 0: K=0,1  1: K=2,3
    const int m    = lane & 15;
    const int mbase = blockIdx.x * 16;        // N_ENT = 6250*16 exactly
    const int obase = wave * 16;

    const float* arow = x + (long)(mbase + m) * D;
    const float* brow = W + (long)(obase + m) * D;   // B[K,N] = W[obase+N, K] (transposed use)

    v8f c = {};
#pragma unroll
    for (int k = 0; k < D; k += 4) {
        v2f a = *(const v2f*)(arow + k + 2 * half);
        v2f b = *(const v2f*)(brow + k + 2 * half);
        c = __builtin_amdgcn_wmma_f32_16x16x4_f32(false, a, false, b,
                                                  (short)0, c, false, false);
    }
#pragma unroll
    for (int j = 0; j < 8; ++j) {
        int row = mbase + j + 8 * half;       // C/D layout: VGPR j -> M = j + 8*half
        emb[(long)row * D + obase + m] = c[j];
    }
}

// ---------------------------------------------------------------------------
// Kernel 2: edge aggregation in D_IN space (linearity refactor):
//   agg[rl, row, :] += w_e * tail_rep[col, :]
// One wave per edge; each lane handles a float4 (32*4 = 128 cols).
// ---------------------------------------------------------------------------
__global__ __launch_bounds__(256) void edge_agg(const float* __restrict__ ew,
                                                const int*   __restrict__ erow,
                                                const int*   __restrict__ ecol,
                                                const float* __restrict__ tail,
                                                float* __restrict__ agg,
                                                int r_start, long aggRelStride) {
    const long e   = (long)blockIdx.x * 8 + (threadIdx.x >> 5); // wave id == edge id
    const int lane = threadIdx.x & 31;
    const int rl   = (int)(e / E_EDGE);
    const int idx  = (int)(e % E_EDGE);
    const long gi  = (long)(r_start + rl) * E_EDGE + idx;

    const float w  = ew[gi];
    const int row  = erow[gi];
    const int col  = ecol[gi];

    const float4 v = *(const float4*)(tail + (long)col * D + lane * 4);
    float* dst = agg + (long)rl * aggRelStride + (long)row * D + lane * 4;
    atomic_add_f32(dst + 0, w * v.x);
    atomic_add_f32(dst + 1, w * v.y);
    atomic_add_f32(dst + 2, w * v.z);
    atomic_add_f32(dst + 3, w * v.w);
}

// ---------------------------------------------------------------------------
// Kernel 3: new_rep[r] = agg[r] @ rel_W[r]  ([B,128] x [128,128]), fused with
// scatter-add: emb[head[r,b], :] += new_rep[r,b,:]  (atomics: cross-relation
// head ids may collide). fp32 WMMA.
// ---------------------------------------------------------------------------
__global__ __launch_bounds__(256) void gemm_agg_scatter(const float* __restrict__ agg,
                                                        const float* __restrict__ relW,
                                                        const int*   __restrict__ head,
                                                        float* __restrict__ emb,
                                                        int r_start, long aggRelStride) {
    const int btilesPerRel = B_HEAD / 16;     // 2048, exact
    const int rl    = blockIdx.x / btilesPerRel;
    const int btile = blockIdx.x % btilesPerRel;
    const int r     = r_start + rl;

    const int wave = threadIdx.x >> 5;
    const int lane = threadIdx.x & 31;
    const int half = lane >> 4;
    const int m    = lane & 15;
    const int bbase = btile * 16;
    const int obase = wave * 16;

    const float* A  = agg + (long)rl * aggRelStride + (long)(bbase + m) * D;
    const float* Wr = relW + (long)r * D * D;          // [in, out] row-major

    v8f c = {};
#pragma unroll
    for (int k = 0; k < D; k += 4) {
        v2f a = *(const v2f*)(A + k + 2 * half);
        const int kb = k + 2 * half;
        v2f b;
        b.x = Wr[(long)kb * D + obase + m];            // B[K,N] = rel_W[k+K, obase+N]
        b.y = Wr[(long)(kb + 1) * D + obase + m];
        c = __builtin_amdgcn_wmma_f32_16x16x4_f32(false, a, false, b,
                                                  (short)0, c, false, false);
    }

    const int* hptr = head + (long)r * B_HEAD;
#pragma unroll
    for (int j = 0; j < 8; ++j) {
        const int localrow = bbase + j + 8 * half;
        const int g = hptr[localrow];
        atomic_add_f32(emb + (long)g * D + obase + m, c[j]);
    }
}

// ---------------------------------------------------------------------------
// Kernel 4: in-place ReLU on d_out (float4 vectorized).
// ---------------------------------------------------------------------------
__global__ __launch_bounds__(256) void relu_inplace(float* __restrict__ p, long n4) {
    const long i = (long)blockIdx.x * 256 + threadIdx.x;
    if (i < n4) {
        float4 v = ((float4*)p)[i];
        v.x = fmaxf(v.x, 0.f);
        v.y = fmaxf(v.y, 0.f);
        v.z = fmaxf(v.z, 0.f);
        v.w = fmaxf(v.w, 0.f);
        ((float4*)p)[i] = v;
    }
}

extern "C" void kernel_launch(void* const* d_in, const int* in_sizes, int n_in,
                              void* d_out, int out_size, void* d_ws, size_t ws_size,
                              hipStream_t stream) {
    const float* x    = (const float*)d_in[0];
    const float* tail = (const float*)d_in[1];
    const float* W    = (const float*)d_in[2];
    const float* relW = (const float*)d_in[3];
    const float* ew   = (const float*)d_in[4];
    const int*   erow = (const int*)d_in[5];
    const int*   ecol = (const int*)d_in[6];
    const int*   head = (const int*)d_in[7];
    float* emb = (float*)d_out;               // pre-ReLU accumulator lives in d_out
    float* agg = (float*)d_ws;

    const size_t aggOne = (size_t)B_HEAD * D * sizeof(float);   // 16.8 MB / relation
    const bool full = ws_size >= aggOne * R_REL;

    // emb = x @ W^T  (fully overwrites d_out)
    gemm_xWt<<<N_ENT / 16, 256, 0, stream>>>(x, W, emb);

    if (full) {
        hipMemsetAsync(agg, 0, aggOne * R_REL, stream);
        edge_agg<<<(R_REL * E_EDGE) / 8, 256, 0, stream>>>(
            ew, erow, ecol, tail, agg, 0, (long)B_HEAD * D);
        gemm_agg_scatter<<<R_REL * (B_HEAD / 16), 256, 0, stream>>>(
            agg, relW, head, emb, 0, (long)B_HEAD * D);
    } else {
        for (int r = 0; r < R_REL; ++r) {
            hipMemsetAsync(agg, 0, aggOne, stream);
            edge_agg<<<E_EDGE / 8, 256, 0, stream>>>(
                ew, erow, ecol, tail, agg, r, 0);
            gemm_agg_scatter<<<B_HEAD / 16, 256, 0, stream>>>(
                agg, relW, head, emb, r, 0);
        }
    }

    relu_inplace<<<((long)N_ENT * D / 4 + 255) / 256, 256, 0, stream>>>(
        emb, (long)N_ENT * D / 4);
}